// IDWT_30382598652219
// MI455X (gfx1250) — compile-verified
//
#include <hip/hip_runtime.h>

// 6-level inverse DWT (db4 synthesis) on (128, n) f32 batches.
// out[j] = sum_i a[i]*rec_lo[j+6-2i] + d[i]*rec_hi[j+6-2i]  (taps in [0,8))
// Per 16(batch)x16(output) tile: D = Xa*Wlo + Xd*Whi with K=12 band,
// executed as 3 K=4 slices x 2 filters = 6x v_wmma_f32_16x16x4_f32 (fp32).
// Weight fragments are tile-invariant -> hoisted; each wave walks 4 tiles.
// Group-level scalarized fast path: no guards, batched b64 loads.

typedef __attribute__((ext_vector_type(2))) float v2f;
typedef __attribute__((ext_vector_type(8))) float v8f;
// Unaligned-capable vector load type (row strides may be odd -> only 4B align).
typedef v2f __attribute__((aligned(4))) v2f_u;

#define TPW 4  // j-tiles per wave

__constant__ float c_rec_lo[8] = {
    0.23037781330885523f,  0.7148465705525415f,  0.6308807679295904f,
   -0.02798376941698385f, -0.18703481171888114f, 0.030841381835986965f,
    0.032883011666982945f,-0.010597401784997278f};
__constant__ float c_rec_hi[8] = {
   -0.010597401784997278f,-0.032883011666982945f, 0.030841381835986965f,
    0.18703481171888114f, -0.02798376941698385f, -0.6308807679295904f,
    0.7148465705525415f,  -0.23037781330885523f};

__device__ __forceinline__ float wtap(const float* w, int t) {
  // w[t] for t in [0,8), else 0. Clamped address, select after the load.
  float v = w[t & 7];
  return (t >= 0 && t < 8) ? v : 0.0f;
}

__global__ __launch_bounds__(128)
void idwt_level_wmma(const float* __restrict__ xa, int stride_a,
                     const float* __restrict__ xd,
                     float* __restrict__ out,
                     int n_in, int L, int j_tiles) {
  const int wave = threadIdx.x >> 5;          // 4 waves per block
  const int lane = threadIdx.x & 31;
  const int wid  = blockIdx.x * 4 + wave;
  const int jt0  = wid * TPW;
  if (__builtin_amdgcn_readfirstlane(jt0 >= j_tiles)) return;  // scalar exit
  const int bt = blockIdx.y;                  // batch tile (128/16 = 8)

  const int m  = lane & 15;                   // A: M row / B,store: N col
  const int hi = lane >> 4;                   // lane-half -> K pair
  const int b0 = bt << 4;

  const float* rowA = xa + (size_t)(b0 + m) * (size_t)stride_a;
  const float* rowD = xd + (size_t)(b0 + m) * (size_t)n_in;

  // B fragments: B[k][n] = rec[n + 6 - 2k], tile-invariant. Build once.
  v2f blo[3], bhi[3];
#pragma unroll
  for (int ks = 0; ks < 3; ++ks) {
    const int kb = ks * 4 + hi * 2;
    const int t0 = m + 6 - 2 * kb;
    blo[ks].x = wtap(c_rec_lo, t0);  blo[ks].y = wtap(c_rec_lo, t0 - 2);
    bhi[ks].x = wtap(c_rec_hi, t0);  bhi[ks].y = wtap(c_rec_hi, t0 - 2);
  }

  // Whole 4-tile group interior, fully stored, no trailing tiles?
  const int grp_fast =
      (jt0 + TPW <= j_tiles) &&
      ((((jt0 + TPW - 1) << 3) + 12) <= n_in) &&
      (((jt0 + TPW) << 4) <= L);

  if (__builtin_amdgcn_readfirstlane(grp_fast)) {
    // ---- Fast path: no guards anywhere. Batch all loads for the group so
    // the scheduler can overlap tile u's WMMAs/stores with tile u+1's loads.
    v2f fa[TPW][3], fd[TPW][3];
#pragma unroll
    for (int u = 0; u < TPW; ++u) {
      const int i0 = (jt0 + u) << 3;
#pragma unroll
      for (int ks = 0; ks < 3; ++ks) {
        const int c = i0 + ks * 4 + hi * 2;
        fa[u][ks] = *(const v2f_u*)(rowA + c);
        fd[u][ks] = *(const v2f_u*)(rowD + c);
      }
    }
#pragma unroll
    for (int u = 0; u < TPW; ++u) {
      v8f acc = {0.f, 0.f, 0.f, 0.f, 0.f, 0.f, 0.f, 0.f};
#pragma unroll
      for (int ks = 0; ks < 3; ++ks) {
        acc = __builtin_amdgcn_wmma_f32_16x16x4_f32(
            false, fa[u][ks], false, blo[ks], (short)0, acc, false, false);
        acc = __builtin_amdgcn_wmma_f32_16x16x4_f32(
            false, fd[u][ks], false, bhi[ks], (short)0, acc, false, false);
      }
      const int j  = ((jt0 + u) << 4) + m;     // always < L here
      const int mb = hi * 8;
#pragma unroll
      for (int r = 0; r < 8; ++r)
        out[(size_t)(b0 + mb + r) * (size_t)L + j] = acc[r];
    }
    return;
  }

  // ---- Edge group (last group of the level only): fully guarded.
  for (int u = 0; u < TPW; ++u) {
    const int jt = jt0 + u;
    if (__builtin_amdgcn_readfirstlane(jt >= j_tiles)) break;
    const int i0 = jt << 3;                   // input band = [8T, 8T+11]
    v8f acc = {0.f, 0.f, 0.f, 0.f, 0.f, 0.f, 0.f, 0.f};
#pragma unroll
    for (int ks = 0; ks < 3; ++ks) {
      const int col0 = i0 + ks * 4 + hi * 2;
      const int col1 = col0 + 1;
      const int cc0 = (col0 < n_in) ? col0 : (n_in - 1);
      const int cc1 = (col1 < n_in) ? col1 : (n_in - 1);
      float a0 = rowA[cc0], a1 = rowA[cc1];
      float d0 = rowD[cc0], d1 = rowD[cc1];
      v2f fa, fd;
      fa.x = (col0 < n_in) ? a0 : 0.f;  fa.y = (col1 < n_in) ? a1 : 0.f;
      fd.x = (col0 < n_in) ? d0 : 0.f;  fd.y = (col1 < n_in) ? d1 : 0.f;
      acc = __builtin_amdgcn_wmma_f32_16x16x4_f32(
          false, fa, false, blo[ks], (short)0, acc, false, false);
      acc = __builtin_amdgcn_wmma_f32_16x16x4_f32(
          false, fd, false, bhi[ks], (short)0, acc, false, false);
    }
    const int j  = (jt << 4) + m;
    const int mb = hi * 8;
    if (j < L) {
#pragma unroll
      for (int r = 0; r < 8; ++r)
        out[(size_t)(b0 + mb + r) * (size_t)L + j] = acc[r];
    }
  }
}

static inline void launch_level(const float* xa, int stride_a, const float* xd,
                                float* out, int n_in, int L, hipStream_t s) {
  const int j_tiles = (L + 15) / 16;
  const int wgroups = (j_tiles + TPW - 1) / TPW;   // waves needed along j
  dim3 grid((wgroups + 3) / 4, 8);                 // 4 waves/block, 8 batch tiles
  idwt_level_wmma<<<grid, 128, 0, s>>>(xa, stride_a, xd, out, n_in, L, j_tiles);
}

extern "C" void kernel_launch(void* const* d_in, const int* in_sizes, int n_in_cnt,
                              void* d_out, int out_size, void* d_ws, size_t ws_size,
                              hipStream_t stream) {
  (void)in_sizes; (void)n_in_cnt; (void)out_size; (void)ws_size;
  const float* a  = (const float*)d_in[0];   // (128, 4102)
  const float* d6 = (const float*)d_in[1];   // (128, 4102)
  const float* d5 = (const float*)d_in[2];   // (128, 8198)
  const float* d4 = (const float*)d_in[3];   // (128, 16390)
  const float* d3 = (const float*)d_in[4];   // (128, 32774)
  const float* d2 = (const float*)d_in[5];   // (128, 65541)
  const float* d1 = (const float*)d_in[6];   // (128, 131075)
  float* out = (float*)d_out;                // (128, 262144)
  float* ws  = (float*)d_ws;                 // needs 131076*128*4 = 67.1 MB

  // Ping-pong ws <-> d_out; intermediates in d_out are strictly smaller than
  // the final output and fully overwritten by the last level.
  launch_level(a,    4102,   d6, ws,   4102,   8198,   stream);
  launch_level(ws,   8198,   d5, out,  8198,   16390,  stream);
  launch_level(out,  16390,  d4, ws,   16390,  32774,  stream);
  launch_level(ws,   32774,  d3, out,  32774,  65542,  stream);
  launch_level(out,  65542,  d2, ws,   65541,  131076, stream);  // trim 65542->65541
  launch_level(ws,   131076, d1, out,  131075, 262144, stream);  // trim 131076->131075
}